// SHEncoder_1047972021050
// MI455X (gfx1250) — compile-verified
//
#include <hip/hip_runtime.h>
#include <hip/hip_bf16.h>
#include <stdint.h>

#define AS1 __attribute__((address_space(1)))
#define AS3 __attribute__((address_space(3)))

typedef int v4i __attribute__((ext_vector_type(4)));
typedef float v4f __attribute__((ext_vector_type(4)));

// 256 threads/block = 8 wave32s. Each block processes 256 points.
// Input staged through LDS via CDNA5 async global->LDS DMA (128-bit, coalesced),
// compute in f32 VALU, output written as 4x nontemporal b128 stores per thread
// (256 MB streaming output > 192 MB L2, so NT stores avoid L2 thrash).
__global__ __launch_bounds__(256) void sh_encode_deg4_kernel(
    const float* __restrict__ dir, float* __restrict__ out, int npts) {
  __shared__ float smem[768];  // 256 points * 3 floats = 3072 B
  const int tid = threadIdx.x;
  const long long base = (long long)blockIdx.x * 256;

  float x, y, z;
  if (base + 256 <= (long long)npts) {
    // Full block: async-stage 3072 B with 192 x b128 DMA transfers.
    if (tid < 192) {
      const char* gsrc = (const char*)dir + base * 12ll + (long long)tid * 16;
      char* ldst = (char*)smem + tid * 16;
#if __has_builtin(__builtin_amdgcn_global_load_async_to_lds_b128)
      __builtin_amdgcn_global_load_async_to_lds_b128(
          (AS1 v4i*)gsrc, (AS3 v4i*)ldst, /*offset=*/0, /*cpol=*/0);
#else
      *(float4*)ldst = *(const float4*)gsrc;   // fallback: plain b128 load+ds store
#endif
    }
#if __has_builtin(__builtin_amdgcn_s_wait_asynccnt)
    __builtin_amdgcn_s_wait_asynccnt(0);
#else
    asm volatile("s_wait_asynccnt 0" ::: "memory");
#endif
    __syncthreads();
    x = smem[3 * tid + 0];
    y = smem[3 * tid + 1];
    z = smem[3 * tid + 2];
  } else {
    // Tail block (not hit for N=4M, kept for safety): direct guarded loads.
    const long long i = base + tid;
    if (i >= npts) return;
    x = __builtin_nontemporal_load(dir + 3 * i + 0);
    y = __builtin_nontemporal_load(dir + 3 * i + 1);
    z = __builtin_nontemporal_load(dir + 3 * i + 2);
  }

  const long long i = base + tid;
  if (i >= npts) return;

  const float x2 = x * x, y2 = y * y, z2 = z * z;
  const float xy = x * y, yz = y * z, xz = x * z;

  v4f r0, r1, r2, r3;
  // l = 0, 1
  r0.x = 0.28209479177387814f;
  r0.y = -0.48860251190291987f * y;
  r0.z =  0.48860251190291987f * z;
  r0.w = -0.48860251190291987f * x;
  // l = 2
  r1.x =  1.0925484305920792f * xy;
  r1.y = -1.0925484305920792f * yz;
  r1.z =  0.94617469575755997f * z2 - 0.31539156525251999f;
  r1.w = -1.0925484305920792f * xz;
  r2.x =  0.54627421529603959f * (x2 - y2);
  // l = 3
  r2.y =  0.59004358992664352f * y * (y2 - 3.0f * x2);
  r2.z =  2.8906114426405538f * xy * z;
  r2.w =  0.45704579946446572f * y * (1.0f - 5.0f * z2);
  r3.x =  0.3731763325901154f  * z * (5.0f * z2 - 3.0f);
  r3.y =  0.45704579946446572f * x * (1.0f - 5.0f * z2);
  r3.z =  1.4453057213202769f  * z * (x2 - y2);
  r3.w =  0.59004358992664352f * x * (3.0f * y2 - x2);

  // Streaming output: nontemporal b128 stores (th:TH_STORE_NT) — written once,
  // never re-read; don't displace L2 contents.
  v4f* o = (v4f*)(out + i * 16);
  __builtin_nontemporal_store(r0, o + 0);
  __builtin_nontemporal_store(r1, o + 1);
  __builtin_nontemporal_store(r2, o + 2);
  __builtin_nontemporal_store(r3, o + 3);
}

extern "C" void kernel_launch(void* const* d_in, const int* in_sizes, int n_in,
                              void* d_out, int out_size, void* d_ws, size_t ws_size,
                              hipStream_t stream) {
  const float* dir = (const float*)d_in[0];   // [npts, 3] f32, normalized
  float* out = (float*)d_out;                 // [npts, 16] f32
  const int npts = in_sizes[0] / 3;
  const int blocks = (npts + 255) / 256;
  sh_encode_deg4_kernel<<<blocks, 256, 0, stream>>>(dir, out, npts);
}